// SSIM_2138893713910
// MI455X (gfx1250) — compile-verified
//
#include <hip/hip_runtime.h>
#include <hip/hip_bf16.h>

// -------------------------------------------------------------------------
// SSIM (fp32, 11-tap separable Gaussian) on gfx1250 using V_WMMA_F32_16X16X4_F32.
// Separable conv expressed as band-matrix GEMM so the fp32 WMMA pipe does the
// filtering while staying in full precision (required for E[x^2]-mu^2).
// One wave (32 threads) per 16x16 output tile; 5 signal planes staged in LDS.
// Band-matrix fragments come from a zero-padded LDS tap table (no select chains).
// -------------------------------------------------------------------------

typedef float v2f __attribute__((ext_vector_type(2)));
typedef float v8f __attribute__((ext_vector_type(8)));

#define WMMA_F32X4(A, B, C) \
  __builtin_amdgcn_wmma_f32_16x16x4_f32(false, (A), false, (B), (short)0, (C), false, false)

namespace {
constexpr int IMG   = 512;
constexpr int OUT   = 502;          // 512 - 11 + 1
constexpr int TAPS  = 11;
constexpr int TILE  = 16;
constexpr int KPAD  = 28;           // 16 + 10 rounded up to multiple of 4 (7 K=4 chunks)
constexpr int NCHUNK= 7;
constexpr int NPLANES = 32 * 3;
constexpr int NTILES  = (OUT + TILE - 1) / TILE;   // 32
constexpr int GPAD_OFF  = 15;       // gpad[t + 15] = g[t]; index kb - lm + 15 in [0, 42]
constexpr int GPAD_SIZE = 48;
constexpr float C1 = 0.01f * 0.01f;
constexpr float C2 = 0.03f * 0.03f;
}

__global__ void __launch_bounds__(32) ssim_zero_kernel(float* __restrict__ out) {
  if (threadIdx.x == 0) out[0] = 0.0f;
}

__global__ void __launch_bounds__(32)
ssim_wmma_kernel(const float* __restrict__ x, const float* __restrict__ y,
                 const float* __restrict__ kern, float* __restrict__ out) {
  // 5 signal planes (x, y, xx, yy, xy), 28 rows x 32 cols each (cols 28..31 = 0).
  __shared__ float sig[5][KPAD][32];
  __shared__ float tmp[TILE][32];     // vertically-convolved 16 x 28 (padded to 32)
  __shared__ float gpad[GPAD_SIZE];   // zero-padded gaussian taps

  const int lane = threadIdx.x;       // wave32
  const int hi   = lane >> 4;         // 0: lanes 0-15, 1: lanes 16-31
  const int lm   = lane & 15;

  const int plane = blockIdx.z;                     // (n, c) fused: 0..95
  const int r0 = blockIdx.y * TILE;                 // output-tile origin == input origin (VALID)
  const int c0 = blockIdx.x * TILE;
  const size_t base = (size_t)plane * IMG * IMG;

  // ---- zero-padded tap table: gpad[t + GPAD_OFF] = g[t] for t in [0,10] ----
  for (int idx = lane; idx < GPAD_SIZE; idx += 32) {
    const int t = idx - GPAD_OFF;
    gpad[idx] = (t >= 0 && t < TAPS) ? kern[t] : 0.0f;
  }

  // ---- stage input tile + fused products into LDS ----
  for (int idx = lane; idx < KPAD * 32; idx += 32) {
    const int row = idx >> 5;
    const int col = idx & 31;
    float xv = 0.0f, yv = 0.0f;
    if (col < KPAD) {
      const int rr = min(r0 + row, IMG - 1);        // clamped halo only feeds masked outputs
      const int cc = min(c0 + col, IMG - 1);
      const size_t off = base + (size_t)rr * IMG + cc;
      xv = x[off];
      yv = y[off];
    }
    sig[0][row][col] = xv;
    sig[1][row][col] = yv;
    sig[2][row][col] = xv * xv;
    sig[3][row][col] = yv * yv;
    sig[4][row][col] = xv * yv;
  }
  __syncthreads();

  // ---- band-matrix fragments (shared by vertical-A and horizontal-B) ----
  // Vertical A:  A_i[m,k] = g[(4i+k) - m], lane holds m=lm, k in {kb, kb+1}, kb = 4i + 2*hi.
  // Horizontal B: B_i[k,c] = g[(4i+k) - c], lane holds c=lm, same k pair -> identical values.
  v2f band[NCHUNK];
#pragma unroll
  for (int i = 0; i < NCHUNK; ++i) {
    const int kb = 4 * i + 2 * hi;
    band[i].x = gpad[kb - lm + GPAD_OFF];
    band[i].y = gpad[kb + 1 - lm + GPAD_OFF];
  }

  v8f res[5];

#pragma unroll
  for (int s = 0; s < 5; ++s) {
    // ---- vertical pass: D = band(16x28) * sig(28x{16|16}) ----
    v8f accL = {}, accR = {};
#pragma unroll
    for (int i = 0; i < NCHUNK; ++i) {
      const int r = 4 * i + 2 * hi;                 // B: K = j + 2*hi, row index into sig
      v2f bL, bR;
      bL.x = sig[s][r][lm];       bL.y = sig[s][r + 1][lm];
      bR.x = sig[s][r][lm + 16];  bR.y = sig[s][r + 1][lm + 16];
      accL = WMMA_F32X4(band[i], bL, accL);
      accR = WMMA_F32X4(band[i], bR, accR);
    }

    // spill D (M = v + 8*hi, N = lm) -> LDS so it can be re-read in A layout
#pragma unroll
    for (int v = 0; v < 8; ++v) {
      tmp[v + 8 * hi][lm]      = accL[v];
      tmp[v + 8 * hi][lm + 16] = accR[v];
    }
    __syncthreads();

    // ---- horizontal pass: D = tmp(16x28) * band(28x16) ----
    v8f acc = {};
#pragma unroll
    for (int i = 0; i < NCHUNK; ++i) {
      const int kb = 4 * i + 2 * hi;                // A: m = lm, k pair at kb
      v2f aF;
      aF.x = tmp[lm][kb];
      aF.y = tmp[lm][kb + 1];
      acc = WMMA_F32X4(aF, band[i], acc);
    }
    res[s] = acc;
    __syncthreads();                                // tmp reused by next signal
  }

  // ---- SSIM map + masked reduction ----
  float lsum = 0.0f;
#pragma unroll
  for (int v = 0; v < 8; ++v) {
    const int orow = r0 + v + 8 * hi;               // D layout: M = v + 8*hi
    const int ocol = c0 + lm;
    const float mu_x = res[0][v];
    const float mu_y = res[1][v];
    const float exx  = res[2][v];
    const float eyy  = res[3][v];
    const float exy  = res[4][v];
    const float mu_xx = mu_x * mu_x;
    const float mu_yy = mu_y * mu_y;
    const float mu_xy = mu_x * mu_y;
    const float s_xx = exx - mu_xx;
    const float s_yy = eyy - mu_yy;
    const float s_xy = exy - mu_xy;
    const float cs = (2.0f * s_xy + C2) / (s_xx + s_yy + C2);
    const float ss = (2.0f * mu_xy + C1) / (mu_xx + mu_yy + C1) * cs;
    lsum += (orow < OUT && ocol < OUT) ? ss : 0.0f;
  }

  // wave32 reduction
#pragma unroll
  for (int off = 16; off > 0; off >>= 1)
    lsum += __shfl_xor(lsum, off, 32);

  if (lane == 0) {
    constexpr float inv_total =
        1.0f / (32.0f * 3.0f * (float)OUT * (float)OUT);
    atomicAdd(out, lsum * inv_total);
  }
}

extern "C" void kernel_launch(void* const* d_in, const int* in_sizes, int n_in,
                              void* d_out, int out_size, void* d_ws, size_t ws_size,
                              hipStream_t stream) {
  (void)in_sizes; (void)n_in; (void)out_size; (void)d_ws; (void)ws_size;
  const float* x    = (const float*)d_in[0];   // (32, 3, 512, 512) fp32
  const float* y    = (const float*)d_in[1];   // (32, 3, 512, 512) fp32
  const float* kern = (const float*)d_in[2];   // (3, 1, 11) fp32 — all rows identical
  float* out        = (float*)d_out;           // scalar

  ssim_zero_kernel<<<1, 32, 0, stream>>>(out);

  dim3 grid(NTILES, NTILES, NPLANES);          // 32 x 32 x 96 one-wave workgroups
  dim3 block(32);
  ssim_wmma_kernel<<<grid, block, 0, stream>>>(x, y, kern, out);
}